// VNDGCNN_4174708211804
// MI455X (gfx1250) — compile-verified
//
#include <hip/hip_runtime.h>
#include <hip/hip_bf16.h>
#include <math.h>

typedef __attribute__((ext_vector_type(2))) float v2f;
typedef __attribute__((ext_vector_type(8))) float v8f;

#define KNN   20
#define NPTS  1024
#define NB    8

// ---------------------------------------------------------------- helpers
__device__ __forceinline__ unsigned f32key(float v) {
  unsigned u = __float_as_uint(v);
  return (u & 0x80000000u) ? ~u : (u | 0x80000000u); // order-preserving uint
}

__device__ __forceinline__ unsigned long long shflxor64(unsigned long long v, int m) {
  unsigned lo = (unsigned)__shfl_xor((int)(unsigned)v, m, 32);
  unsigned hi = (unsigned)__shfl_xor((int)(unsigned)(v >> 32), m, 32);
  return ((unsigned long long)hi << 32) | lo;
}

// ---------------------------------------------------------------- col sq-norms
__global__ void colnorm_kernel(const float* __restrict__ X, long bstride, int K,
                               float* __restrict__ xx) {
  int b = blockIdx.y;
  int n = blockIdx.x * blockDim.x + threadIdx.x;
  if (n >= NPTS) return;
  const float* Xb = X + (long)b * bstride;
  float s = 0.f;
  for (int c = 0; c < K; ++c) { float v = Xb[(long)c * NPTS + n]; s += v * v; }
  xx[b * NPTS + n] = s;
}

// ---------------------------------------------------------------- KNN: WMMA Gram + wave32 top-20
// block = 128 (4 waves); grid = (NPTS/16, B); dynamic LDS = 16*NPTS floats
template<int KCH>
__global__ void knn_kernel(const float* __restrict__ X, long bstride,
                           const float* __restrict__ xx, int* __restrict__ idx) {
  extern __shared__ float pd[];            // 16 rows x 1024 cols of -dist^2
  const int b    = blockIdx.y;
  const int row0 = blockIdx.x * 16;
  const float* Xb  = X + (long)b * bstride;
  const float* xxb = xx + b * NPTS;
  const int lane = threadIdx.x & 31;
  const int wave = threadIdx.x >> 5;
  const int l16  = lane & 15;
  const int hi   = lane >> 4;

  // ---- preload all A fragments (row stripe) once; clamp+select, no exec div ----
  constexpr int NKF = (KCH + 3) / 4;
  v2f afrag[NKF];
#pragma unroll
  for (int f = 0; f < NKF; ++f) {
    const int ka  = f * 4 + hi * 2;        // f32 A layout: v0 = K(+0/+2), v1 = K(+1/+3)
    const int ka0 = (ka     < KCH) ? ka     : KCH - 1;
    const int ka1 = (ka + 1 < KCH) ? ka + 1 : KCH - 1;
    const float a0 = Xb[(long)ka0 * NPTS + row0 + l16];
    const float a1 = Xb[(long)ka1 * NPTS + row0 + l16];
    afrag[f].x = (ka     < KCH) ? a0 : 0.f;
    afrag[f].y = (ka + 1 < KCH) ? a1 : 0.f;
  }

  // ---- Gram tiles via V_WMMA_F32_16X16X4_F32 ----
  for (int ct = wave; ct < NPTS / 16; ct += 4) {
    const int col0 = ct * 16;
    v8f acc = {};
#pragma unroll
    for (int f = 0; f < NKF; ++f) {
      const int ka  = f * 4 + hi * 2;
      const int ka0 = (ka     < KCH) ? ka     : KCH - 1;
      const int ka1 = (ka + 1 < KCH) ? ka + 1 : KCH - 1;
      const float b0 = Xb[(long)ka0 * NPTS + col0 + l16];
      const float b1 = Xb[(long)ka1 * NPTS + col0 + l16];
      v2f bm;
      bm.x = (ka     < KCH) ? b0 : 0.f;
      bm.y = (ka + 1 < KCH) ? b1 : 0.f;
      acc = __builtin_amdgcn_wmma_f32_16x16x4_f32(false, afrag[f], false, bm,
                                                  (short)0, acc, false, false);
    }
    const int   j  = col0 + l16;
    const float xj = xxb[j];
#pragma unroll
    for (int r = 0; r < 8; ++r) {
      const int m = r + 8 * hi;            // C/D layout: vgpr r -> row r (+8 for hi lanes)
      pd[m * NPTS + j] = 2.f * acc[r] - xxb[row0 + m] - xj;   // pd = -||xi-xj||^2
    }
  }
  __syncthreads();

  // ---- top-20 per row: lane-local sorted lists + butterfly argmax merge ----
  for (int rr = 0; rr < 4; ++rr) {
    const int m = wave * 4 + rr;
    float vals[KNN]; int cols[KNN];
#pragma unroll
    for (int t = 0; t < KNN; ++t) { vals[t] = -3.4e38f; cols[t] = 0x7FFFFFFF; }
    for (int j = lane; j < NPTS; j += 32) {
      const float v = pd[m * NPTS + j];
      if (v > vals[KNN - 1]) {
        int t = KNN - 1;
        while (t > 0 && vals[t - 1] < v) { vals[t] = vals[t - 1]; cols[t] = cols[t - 1]; --t; }
        vals[t] = v; cols[t] = j;
      }
    }
    int p = 0;
    int* orow = idx + ((long)b * NPTS + row0 + m) * KNN;
    for (int t = 0; t < KNN; ++t) {
      const unsigned long long key =
          ((unsigned long long)f32key(vals[p]) << 32) |
          (0xFFFFFFFFull - (unsigned)cols[p]);               // tie -> smaller index
      unsigned long long best = key;
#pragma unroll
      for (int s = 16; s >= 1; s >>= 1) {
        unsigned long long o = shflxor64(best, s);
        if (o > best) best = o;
      }
      if (best == key) ++p;                                  // unique winner pops
      if (lane == 0) orow[t] = (int)(0xFFFFFFFFu - (unsigned)(best & 0xFFFFFFFFull));
    }
  }
}

// ---------------------------------------------------------------- misc
__global__ void zero_kernel(float* p, int n) {
  int i = blockIdx.x * blockDim.x + threadIdx.x;
  if (i < n) p[i] = 0.f;
}

// ---------------------------------------------------------------- pass A: BN statistics
// grid = (NPTS, B); block = 256; dyn LDS = (KNN+1)*Cin*3 + 2*Cout floats
template<int Cin, int Cout>
__global__ void passA_kernel(const float* __restrict__ xin, long bstride,
                             const float* __restrict__ W, const int* __restrict__ idx,
                             float* __restrict__ sum, float* __restrict__ ssq) {
  extern __shared__ float sm[];
  const int b = blockIdx.y, n = blockIdx.x;
  const float* xb = xin + (long)b * bstride;
  float* feat = sm;                         // [KNN+1][Cin*3] (slot KNN = center)
  float* ls   = feat + (KNN + 1) * Cin * 3; // [Cout]
  float* lq   = ls + Cout;                  // [Cout]
  const int* ib = idx + ((long)b * NPTS + n) * KNN;
  const int tid = threadIdx.x, nthr = blockDim.x;

  for (int t = tid; t < (KNN + 1) * Cin * 3; t += nthr) {
    const int k = t / (Cin * 3), cv = t % (Cin * 3);
    const int src = (k == KNN) ? n : ib[k];
    feat[t] = xb[(long)cv * NPTS + src];
  }
  for (int t = tid; t < Cout; t += nthr) { ls[t] = 0.f; lq[t] = 0.f; }
  __syncthreads();

  for (int t = tid; t < Cout * KNN; t += nthr) {
    const int o = t / KNN, k = t % KNN;
    const float* wr = W + o * (2 * Cin);
    const float* fk = feat + k * Cin * 3;
    const float* fc = feat + KNN * Cin * 3;
    float p0 = 0.f, p1 = 0.f, p2 = 0.f;
#pragma unroll 4
    for (int c = 0; c < Cin; ++c) {
      const float w1 = wr[c], w2 = wr[Cin + c];
      const float c0 = fc[c*3+0], c1 = fc[c*3+1], c2 = fc[c*3+2];
      p0 += w1 * (fk[c*3+0] - c0) + w2 * c0;
      p1 += w1 * (fk[c*3+1] - c1) + w2 * c1;
      p2 += w1 * (fk[c*3+2] - c2) + w2 * c2;
    }
    const float nm = sqrtf(p0*p0 + p1*p1 + p2*p2) + 1e-6f;
    atomicAdd(&ls[o], nm);
    atomicAdd(&lq[o], nm * nm);
  }
  __syncthreads();
  for (int t = tid; t < Cout; t += nthr) {
    atomicAdd(&sum[t], ls[t]);
    atomicAdd(&ssq[t], lq[t]);
  }
}

// ---------------------------------------------------------------- pass B: apply + VN-LeakyReLU + VN-maxpool (fused)
// grid = (NPTS, B); block = 256; dyn LDS = (KNN+1)*Cin*3 + Cout*3*KNN + Cout*KNN floats
template<int Cin, int Cout>
__global__ void passB_kernel(const float* __restrict__ xin, long bstride,
                             const float* __restrict__ W, const float* __restrict__ D,
                             const float* __restrict__ g, const float* __restrict__ bet,
                             const float* __restrict__ M, const int* __restrict__ idx,
                             const float* __restrict__ sum, const float* __restrict__ ssq,
                             float* __restrict__ xout, long out_bstride, int out_c0) {
  extern __shared__ float sm[];
  const int b = blockIdx.y, n = blockIdx.x;
  const float* xb = xin + (long)b * bstride;
  float* feat = sm;                          // [KNN+1][Cin*3]
  float* h    = feat + (KNN + 1) * Cin * 3;  // [Cout*3][KNN]
  float* dt   = h + Cout * 3 * KNN;          // [Cout][KNN]
  const int* ib = idx + ((long)b * NPTS + n) * KNN;
  const int tid = threadIdx.x, nthr = blockDim.x;
  const float cnt = (float)(NB * NPTS * KNN);

  for (int t = tid; t < (KNN + 1) * Cin * 3; t += nthr) {
    const int k = t / (Cin * 3), cv = t % (Cin * 3);
    const int src = (k == KNN) ? n : ib[k];
    feat[t] = xb[(long)cv * NPTS + src];
  }
  __syncthreads();

  for (int t = tid; t < Cout * KNN; t += nthr) {
    const int o = t / KNN, k = t % KNN;
    const float* wr = W + o * (2 * Cin);
    const float* dr = D + o * (2 * Cin);
    const float* fk = feat + k * Cin * 3;
    const float* fc = feat + KNN * Cin * 3;
    float p0 = 0.f, p1 = 0.f, p2 = 0.f, d0 = 0.f, d1 = 0.f, d2 = 0.f;
#pragma unroll 4
    for (int c = 0; c < Cin; ++c) {
      const float c0 = fc[c*3+0], c1 = fc[c*3+1], c2 = fc[c*3+2];
      const float e0 = fk[c*3+0] - c0, e1 = fk[c*3+1] - c1, e2 = fk[c*3+2] - c2;
      const float w1 = wr[c], w2 = wr[Cin + c];
      const float u1 = dr[c], u2 = dr[Cin + c];
      p0 += w1 * e0 + w2 * c0;  p1 += w1 * e1 + w2 * c1;  p2 += w1 * e2 + w2 * c2;
      d0 += u1 * e0 + u2 * c0;  d1 += u1 * e1 + u2 * c1;  d2 += u1 * e2 + u2 * c2;
    }
    const float nm   = sqrtf(p0*p0 + p1*p1 + p2*p2) + 1e-6f;
    const float mean = sum[o] / cnt;
    const float var  = ssq[o] / cnt - mean * mean;
    const float nbn  = (nm - mean) * rsqrtf(var + 1e-5f) * g[o] + bet[o];
    const float sc   = nbn / nm;
    p0 *= sc; p1 *= sc; p2 *= sc;
    const float dot = p0*d0 + p1*d1 + p2*d2;
    const float dsq = d0*d0 + d1*d1 + d2*d2;
    const float f   = dot / (dsq + 1e-6f);
    const bool  pos = (dot >= 0.f);
    h[(o*3+0)*KNN + k] = 0.2f*p0 + 0.8f*(pos ? p0 : p0 - f*d0);
    h[(o*3+1)*KNN + k] = 0.2f*p1 + 0.8f*(pos ? p1 : p1 - f*d1);
    h[(o*3+2)*KNN + k] = 0.2f*p2 + 0.8f*(pos ? p2 : p2 - f*d2);
  }
  __syncthreads();

  // pool direction: dt[o][k] = sum_v h[o][v][k] * (M h)[o][v][k]
  for (int t = tid; t < Cout * KNN; t += nthr) {
    const int o = t / KNN, k = t % KNN;
    const float* mr = M + o * Cout;
    float a0 = 0.f, a1 = 0.f, a2 = 0.f;
#pragma unroll 4
    for (int c = 0; c < Cout; ++c) {
      const float mv = mr[c];
      a0 += mv * h[(c*3+0)*KNN + k];
      a1 += mv * h[(c*3+1)*KNN + k];
      a2 += mv * h[(c*3+2)*KNN + k];
    }
    dt[o*KNN + k] = h[(o*3+0)*KNN + k]*a0 + h[(o*3+1)*KNN + k]*a1 + h[(o*3+2)*KNN + k]*a2;
  }
  __syncthreads();

  for (int o = tid; o < Cout; o += nthr) {
    int kb = 0; float best = dt[o*KNN];
    for (int k = 1; k < KNN; ++k) { const float v = dt[o*KNN + k]; if (v > best) { best = v; kb = k; } }
    float* ob = xout + (long)b * out_bstride + (long)(out_c0 + o) * 3 * NPTS + n;
    ob[0]        = h[(o*3+0)*KNN + kb];
    ob[NPTS]     = h[(o*3+1)*KNN + kb];
    ob[2*NPTS]   = h[(o*3+2)*KNN + kb];
  }
}

// ---------------------------------------------------------------- layer 5: WMMA GEMM  p = W5 (341x169) * xc (169x3072) per batch
// guard-free main K loop (k=0..167), explicit 1-deep tail (k=168); rows clamped (garbage rows never stored)
__global__ void gemm5_kernel(const float* __restrict__ W5, const float* __restrict__ xc,
                             float* __restrict__ out) {
  const int Kc = 169, Mtot = 341, Jtot = 3072;
  const int b  = blockIdx.z;
  const int m0 = blockIdx.y * 16;
  const int j0 = blockIdx.x * 16;
  const float* xb = xc + (long)b * Kc * Jtot;
  const int lane = threadIdx.x & 31;
  const int l16  = lane & 15;
  const int hi   = lane >> 4;
  const int row  = m0 + l16;
  const int rowc = (row < Mtot) ? row : Mtot - 1;       // clamp, no exec divergence
  const float* wr = W5 + (long)rowc * Kc;
  v8f acc = {};
#pragma unroll 4
  for (int kk = 0; kk < 168; kk += 4) {                 // 42 full, guard-free steps
    const int ka = kk + hi * 2;
    v2f a, bm;
    a.x  = wr[ka];
    a.y  = wr[ka + 1];
    bm.x = xb[(long)ka * Jtot + j0 + l16];
    bm.y = xb[(long)(ka + 1) * Jtot + j0 + l16];
    if (kk + 4 < 168) __builtin_prefetch(&xb[(long)(kk + 4) * Jtot + j0 + l16], 0, 1);
    acc = __builtin_amdgcn_wmma_f32_16x16x4_f32(false, a, false, bm,
                                                (short)0, acc, false, false);
  }
  {                                                     // tail: k = 168 only
    v2f a, bm;
    const float av = wr[Kc - 1];
    const float bv = xb[(long)(Kc - 1) * Jtot + j0 + l16];
    a.x  = (hi == 0) ? av : 0.f;  a.y  = 0.f;
    bm.x = (hi == 0) ? bv : 0.f;  bm.y = 0.f;
    acc = __builtin_amdgcn_wmma_f32_16x16x4_f32(false, a, false, bm,
                                                (short)0, acc, false, false);
  }
#pragma unroll
  for (int r = 0; r < 8; ++r) {
    const int m = m0 + r + 8 * hi;
    if (m < Mtot) out[(long)b * Mtot * Jtot + (long)m * Jtot + j0 + l16] = acc[r];
  }
}

// stats over norm(p) for layer 5 BN; grid = (341, B), block = 256
__global__ void stats5_kernel(const float* __restrict__ p, float* __restrict__ sum,
                              float* __restrict__ ssq) {
  const int o = blockIdx.x, b = blockIdx.y;
  const float* pb = p + ((long)b * 341 + o) * 3072;
  float s = 0.f, s2 = 0.f;
  for (int n = threadIdx.x; n < NPTS; n += blockDim.x) {
    const float p0 = pb[n], p1 = pb[NPTS + n], p2 = pb[2*NPTS + n];
    const float nm = sqrtf(p0*p0 + p1*p1 + p2*p2) + 1e-6f;
    s += nm; s2 += nm * nm;
  }
  __shared__ float rs[256], rq[256];
  rs[threadIdx.x] = s; rq[threadIdx.x] = s2;
  __syncthreads();
  for (int st = 128; st > 0; st >>= 1) {
    if ((int)threadIdx.x < st) { rs[threadIdx.x] += rs[threadIdx.x + st];
                                 rq[threadIdx.x] += rq[threadIdx.x + st]; }
    __syncthreads();
  }
  if (threadIdx.x == 0) { atomicAdd(&sum[o], rs[0]); atomicAdd(&ssq[o], rq[0]); }
}

// d = D5 (1x169) * xc ; grid = (12, B), block = 256
__global__ void dvec_kernel(const float* __restrict__ D5, const float* __restrict__ xc,
                            float* __restrict__ dvec) {
  const int b = blockIdx.y;
  const int j = blockIdx.x * blockDim.x + threadIdx.x;  // j in [0, 3072)
  const float* xb = xc + (long)b * 169 * 3072;
  float s = 0.f;
#pragma unroll 4
  for (int c = 0; c < 169; ++c) s += D5[c] * xb[(long)c * 3072 + j];
  dvec[b * 3072 + j] = s;
}

// final BN + VN-LeakyReLU in place on d_out; grid = (4, 341, B), block = 256
__global__ void apply5_kernel(float* __restrict__ p, const float* __restrict__ dvec,
                              const float* __restrict__ sum, const float* __restrict__ ssq,
                              const float* __restrict__ g, const float* __restrict__ bet) {
  const int n = blockIdx.x * blockDim.x + threadIdx.x;
  const int o = blockIdx.y, b = blockIdx.z;
  const float cnt  = (float)(NB * NPTS);
  const float mean = sum[o] / cnt;
  const float var  = ssq[o] / cnt - mean * mean;
  float* pb = p + ((long)b * 341 + o) * 3072;
  float p0 = pb[n], p1 = pb[NPTS + n], p2 = pb[2*NPTS + n];
  const float nm  = sqrtf(p0*p0 + p1*p1 + p2*p2) + 1e-6f;
  const float nbn = (nm - mean) * rsqrtf(var + 1e-5f) * g[o] + bet[o];
  const float sc  = nbn / nm;
  p0 *= sc; p1 *= sc; p2 *= sc;
  const float* db = dvec + b * 3072;
  const float d0 = db[n], d1 = db[NPTS + n], d2 = db[2*NPTS + n];
  const float dot = p0*d0 + p1*d1 + p2*d2;
  const float dsq = d0*d0 + d1*d1 + d2*d2;
  const float f   = dot / (dsq + 1e-6f);
  const bool pos  = (dot >= 0.f);
  pb[n]          = 0.2f*p0 + 0.8f*(pos ? p0 : p0 - f*d0);
  pb[NPTS + n]   = 0.2f*p1 + 0.8f*(pos ? p1 : p1 - f*d1);
  pb[2*NPTS + n] = 0.2f*p2 + 0.8f*(pos ? p2 : p2 - f*d2);
}

// ---------------------------------------------------------------- host
extern "C" void kernel_launch(void* const* d_in, const int* in_sizes, int n_in,
                              void* d_out, int out_size, void* d_ws, size_t ws_size,
                              hipStream_t stream) {
  (void)in_sizes; (void)n_in; (void)out_size; (void)ws_size;
  const float* x  = (const float*)d_in[0];
  const float* W1 = (const float*)d_in[1];  const float* D1 = (const float*)d_in[2];
  const float* g1 = (const float*)d_in[3];  const float* b1 = (const float*)d_in[4];
  const float* M1 = (const float*)d_in[5];
  const float* W2 = (const float*)d_in[6];  const float* D2 = (const float*)d_in[7];
  const float* g2 = (const float*)d_in[8];  const float* b2 = (const float*)d_in[9];
  const float* M2 = (const float*)d_in[10];
  const float* W3 = (const float*)d_in[11]; const float* D3 = (const float*)d_in[12];
  const float* g3 = (const float*)d_in[13]; const float* b3 = (const float*)d_in[14];
  const float* M3 = (const float*)d_in[15];
  const float* W4 = (const float*)d_in[16]; const float* D4 = (const float*)d_in[17];
  const float* g4 = (const float*)d_in[18]; const float* b4 = (const float*)d_in[19];
  const float* M4 = (const float*)d_in[20];
  const float* W5 = (const float*)d_in[21]; const float* D5 = (const float*)d_in[22];
  const float* g5 = (const float*)d_in[23]; const float* b5 = (const float*)d_in[24];
  float* out = (float*)d_out;

  float* ws   = (float*)d_ws;
  float* XX   = ws;                         // 8192
  float* SUM  = ws + 8192;                  // 512
  float* SSQ  = ws + 8704;                  // 512
  float* DVEC = ws + 9216;                  // 24576
  float* XC   = ws + 33792;                 // 8*169*3072 = 4,153,344
  int*   IDX  = (int*)(XC + 4153344L);      // 8*1024*20 ints

  const long XCB = 169L * 3 * NPTS;         // 519168 : batch stride of concat buffer
  const dim3 b256(256);
  const size_t knnLds = 16 * NPTS * sizeof(float);
  auto shA = [](int Cin, int Cout) {
    return (size_t)(((KNN + 1) * Cin * 3 + 2 * Cout) * sizeof(float)); };
  auto shB = [](int Cin, int Cout) {
    return (size_t)(((KNN + 1) * Cin * 3 + Cout * 3 * KNN + Cout * KNN) * sizeof(float)); };

  // ---------------- layer 1 : x (C=1) -> x1 (21ch) at XC[c0=0]
  colnorm_kernel<<<dim3(4, NB), b256, 0, stream>>>(x, 3072L, 3, XX);
  knn_kernel<3><<<dim3(64, NB), dim3(128), knnLds, stream>>>(x, 3072L, XX, IDX);
  zero_kernel<<<dim3(4), b256, 0, stream>>>(SUM, 1024);
  passA_kernel<1, 21><<<dim3(NPTS, NB), b256, shA(1, 21), stream>>>(x, 3072L, W1, IDX, SUM, SSQ);
  passB_kernel<1, 21><<<dim3(NPTS, NB), b256, shB(1, 21), stream>>>(
      x, 3072L, W1, D1, g1, b1, M1, IDX, SUM, SSQ, XC, XCB, 0);

  // ---------------- layer 2 : x1 (21ch) -> x2 (21ch) at XC[c0=21]
  colnorm_kernel<<<dim3(4, NB), b256, 0, stream>>>(XC, XCB, 63, XX);
  knn_kernel<63><<<dim3(64, NB), dim3(128), knnLds, stream>>>(XC, XCB, XX, IDX);
  zero_kernel<<<dim3(4), b256, 0, stream>>>(SUM, 1024);
  passA_kernel<21, 21><<<dim3(NPTS, NB), b256, shA(21, 21), stream>>>(XC, XCB, W2, IDX, SUM, SSQ);
  passB_kernel<21, 21><<<dim3(NPTS, NB), b256, shB(21, 21), stream>>>(
      XC, XCB, W2, D2, g2, b2, M2, IDX, SUM, SSQ, XC, XCB, 21);

  // ---------------- layer 3 : x2 (21ch) -> x3 (42ch) at XC[c0=42]
  const float* x2p = XC + 21L * 3 * NPTS;
  colnorm_kernel<<<dim3(4, NB), b256, 0, stream>>>(x2p, XCB, 63, XX);
  knn_kernel<63><<<dim3(64, NB), dim3(128), knnLds, stream>>>(x2p, XCB, XX, IDX);
  zero_kernel<<<dim3(4), b256, 0, stream>>>(SUM, 1024);
  passA_kernel<21, 42><<<dim3(NPTS, NB), b256, shA(21, 42), stream>>>(x2p, XCB, W3, IDX, SUM, SSQ);
  passB_kernel<21, 42><<<dim3(NPTS, NB), b256, shB(21, 42), stream>>>(
      x2p, XCB, W3, D3, g3, b3, M3, IDX, SUM, SSQ, XC, XCB, 42);

  // ---------------- layer 4 : x3 (42ch) -> x4 (85ch) at XC[c0=84]
  const float* x3p = XC + 42L * 3 * NPTS;
  colnorm_kernel<<<dim3(4, NB), b256, 0, stream>>>(x3p, XCB, 126, XX);
  knn_kernel<126><<<dim3(64, NB), dim3(128), knnLds, stream>>>(x3p, XCB, XX, IDX);
  zero_kernel<<<dim3(4), b256, 0, stream>>>(SUM, 1024);
  passA_kernel<42, 85><<<dim3(NPTS, NB), b256, shA(42, 85), stream>>>(x3p, XCB, W4, IDX, SUM, SSQ);
  passB_kernel<42, 85><<<dim3(NPTS, NB), b256, shB(42, 85), stream>>>(
      x3p, XCB, W4, D4, g4, b4, M4, IDX, SUM, SSQ, XC, XCB, 84);

  // ---------------- layer 5 : vn_lrelu(W5 * xc) -> d_out (8,341,3,1024)
  zero_kernel<<<dim3(4), b256, 0, stream>>>(SUM, 1024);
  gemm5_kernel<<<dim3(192, 22, NB), dim3(32), 0, stream>>>(W5, XC, out);
  stats5_kernel<<<dim3(341, NB), b256, 0, stream>>>(out, SUM, SSQ);
  dvec_kernel<<<dim3(12, NB), b256, 0, stream>>>(D5, XC, DVEC);
  apply5_kernel<<<dim3(4, 341, NB), b256, 0, stream>>>(out, DVEC, SUM, SSQ, g5, b5);
}